// EnvelopeExtractor_55233279427163
// MI455X (gfx1250) — compile-verified
//
#include <hip/hip_runtime.h>
#include <math.h>

// ---------------- compile-time biquad coefficients -------------------------
constexpr double KPI = 3.14159265358979323846;
constexpr double c_sin(double x) {
    double x2 = x * x;
    return x * (1.0 - x2 / 6.0 + x2 * x2 / 120.0 - x2 * x2 * x2 / 5040.0);
}
constexpr double c_cos(double x) {
    double x2 = x * x;
    return 1.0 - x2 / 2.0 + x2 * x2 / 24.0 - x2 * x2 * x2 / 720.0;
}
struct DC { double b0, b1, b2, a1, a2; };
constexpr DC make_dc() {
    double w0 = 2.0 * KPI * 10.0 / 24000.0;
    double alpha = c_sin(w0) / (2.0 * 0.707);
    double cw = c_cos(w0);
    double b0 = (1.0 + cw) / 2.0, b1 = -(1.0 + cw), b2 = b0;
    double a0 = 1.0 + alpha, a1 = -2.0 * cw, a2 = 1.0 - alpha;
    return { b0 / a0, b1 / a0, b2 / a0, a1 / a0, a2 / a0 };
}
constexpr DC DCO = make_dc();
constexpr float B0f = (float)DCO.b0, B1f = (float)DCO.b1, B2f = (float)DCO.b2;
constexpr float A1f = (float)DCO.a1, A2f = (float)DCO.a2;

// ---------------- problem geometry ----------------------------------------
constexpr int BATCH = 64;
constexpr int TLEN  = 480000;
constexpr int LCH   = 640;               // chunk length (640*750 = 480000)
constexpr int CCH   = TLEN / LCH;        // 750 chunks per row
constexpr int BC    = BATCH * CCH;       // 48000 sequences
constexpr int TSTEP = 64;                // time steps per async-staged tile
constexpr int NT    = LCH / TSTEP;       // 10 tiles per chunk
constexpr int ROWF  = 68;                // padded LDS row stride (banks 4n+t)

// ---------------- compile-time recurrence tables ---------------------------
struct Tables {
    float g1[16], g2[16];
    float Ax[4][2][16], Ay[4][2][16];
    float m00, m01, m10, m11;
};
constexpr Tables make_tables() {
    Tables t{};
    double a1 = DCO.a1, a2 = DCO.a2;
    double h[16] = {};
    {
        double p1 = 0.0, p2 = 0.0;
        for (int n = 0; n < 16; ++n) {
            double cur = (n == 0 ? 1.0 : 0.0) - a1 * p1 - a2 * p2;
            h[n] = cur; p2 = p1; p1 = cur;
        }
    }
    {
        double p1 = 1.0, p2 = 0.0;                       // g1: response to y[-1]
        for (int n = 0; n < 16; ++n) {
            double cur = -a1 * p1 - a2 * p2;
            t.g1[n] = (float)cur; p2 = p1; p1 = cur;
        }
    }
    {
        double p1 = 0.0, p2 = 1.0;                       // g2: response to y[-2]
        for (int n = 0; n < 16; ++n) {
            double cur = -a1 * p1 - a2 * p2;
            t.g2[n] = (float)cur; p2 = p1; p1 = cur;
        }
    }
    for (int j = 0; j < 4; ++j)
        for (int hi = 0; hi < 2; ++hi)
            for (int m = 0; m < 16; ++m) {
                int k = 4 * j + 2 * hi;
                t.Ax[j][hi][m] = (m >= k)     ? (float)h[m - k]     : 0.0f;
                t.Ay[j][hi][m] = (m >= k + 1) ? (float)h[m - k - 1] : 0.0f;
            }
    {
        double m00 = 1, m01 = 0, m10 = 0, m11 = 1;       // A^LCH
        for (int i = 0; i < LCH; ++i) {
            double n00 = -a1 * m00 - a2 * m10;
            double n01 = -a1 * m01 - a2 * m11;
            double n10 = m00, n11 = m01;
            m00 = n00; m01 = n01; m10 = n10; m11 = n11;
        }
        t.m00 = (float)m00; t.m01 = (float)m01;
        t.m10 = (float)m10; t.m11 = (float)m11;
    }
    return t;
}
__device__ constexpr Tables GT = make_tables();

typedef float v2f __attribute__((ext_vector_type(2)));
typedef float v8f __attribute__((ext_vector_type(8)));
typedef int   v4i __attribute__((ext_vector_type(4)));
typedef __attribute__((address_space(1))) v4i gv4i;   // global int4
typedef __attribute__((address_space(3))) v4i lv4i;   // LDS int4

// ---------------- gfx1250 async global->LDS helpers ------------------------
#if __has_builtin(__builtin_amdgcn_global_load_async_to_lds_b128) && \
    __has_builtin(__builtin_amdgcn_s_wait_asynccnt)
#define EE_HAVE_ASYNC 1
#else
#define EE_HAVE_ASYNC 0
#endif

__device__ __forceinline__ void ee_copy_b128(const float* g, float* l) {
#if EE_HAVE_ASYNC
    __builtin_amdgcn_global_load_async_to_lds_b128(
        (gv4i*)(v4i*)(void*)const_cast<float*>(g),
        (lv4i*)(v4i*)(void*)l, 0, 0);
#else
    *reinterpret_cast<float4*>(l) = *reinterpret_cast<const float4*>(g);
#endif
}
__device__ __forceinline__ void ee_wait_async_le8() {
#if EE_HAVE_ASYNC
    __builtin_amdgcn_s_wait_asynccnt(8);
#endif
}
__device__ __forceinline__ void ee_wait_async_0() {
#if EE_HAVE_ASYNC
    __builtin_amdgcn_s_wait_asynccnt(0);
#endif
}

// ---------------- phase 1: per-chunk particular final state ----------------
__global__ __launch_bounds__(128) void ee_phase1(const float* __restrict__ x,
                                                 float2* __restrict__ pst) {
    int s = blockIdx.x * 128 + threadIdx.x;
    if (s >= BC) return;
    int b = s / CCH, c = s - b * CCH;
    const float* xp = x + (long)b * TLEN + (long)c * LCH;   // 16B aligned
    float x1 = (c > 0) ? fabsf(xp[-1]) : 0.0f;
    float x2 = (c > 0) ? fabsf(xp[-2]) : 0.0f;
    float y1 = 0.0f, y2 = 0.0f;
    for (int i = 0; i < LCH; i += 32) {
        __builtin_prefetch(xp + i + 512, 0, 0);   // speculative: safe OOB
#pragma unroll
        for (int u = 0; u < 8; ++u) {
            float4 v4 = *reinterpret_cast<const float4*>(xp + i + 4 * u);
            float xs[4] = { v4.x, v4.y, v4.z, v4.w };
#pragma unroll
            for (int q = 0; q < 4; ++q) {
                float xa  = fabsf(xs[q]);
                float fir = B0f * xa + B1f * x1 + B2f * x2;
                float yv  = fir - A1f * y1 - A2f * y2;
                y2 = y1; y1 = yv; x2 = x1; x1 = xa;
            }
        }
    }
    pst[s] = make_float2(y1, y2);
}

// ---------------- phase 2: sequential inter-chunk combine ------------------
__global__ __launch_bounds__(64) void ee_phase2(const float2* __restrict__ p,
                                                float2* __restrict__ ini) {
    int b = threadIdx.x;                 // 64 rows, one lane each
    const float m00 = GT.m00, m01 = GT.m01, m10 = GT.m10, m11 = GT.m11;
    float s1 = 0.0f, s2 = 0.0f;          // (y[-1], y[-2]) entering next chunk
    int base = b * CCH;
    for (int c = 0; c < CCH; ++c) {
        ini[base + c] = make_float2(s1, s2);
        float2 pc = p[base + c];
        float n1 = m00 * s1 + m01 * s2 + pc.x;
        float n2 = m10 * s1 + m11 * s2 + pc.y;
        s1 = n1; s2 = n2;
    }
}

// ---------------- phase 3: async-staged WMMA blocked scan ------------------
__global__ __launch_bounds__(32) void ee_phase3(const float* __restrict__ x,
                                                const float2* __restrict__ ini,
                                                float* __restrict__ y) {
    __shared__ float stage[2][16][ROWF];             // double-buffered x tiles
    __shared__ float lt[16][18];                     // output transpose tile
    const int lane  = threadIdx.x;                   // wave32, EXEC all ones
    const int n     = lane & 15;                     // column / A row / seq
    const int hi    = lane >> 4;                     // K/M half select
    const int wseq0 = blockIdx.x * 16;

    // this lane's compute column (sequence)
    int seqn = wseq0 + n;
    int bn = seqn / CCH, cn = seqn - bn * CCH;
    const float* xcol = x + (long)bn * TLEN + (long)cn * LCH;
    int tstart = cn * LCH;

    // this lane's store column (after LDS transpose)
    int sseq = lane >> 1, th = (lane & 1) * 8;
    int seq2 = wseq0 + sseq;
    int b2 = seq2 / CCH, c2 = seq2 - b2 * CCH;
    float* ycol = y + (long)b2 * TLEN + (long)c2 * LCH;

    // constant operands (loaded once per wave)
    v2f av[4];
#pragma unroll
    for (int j = 0; j < 4; ++j) { av[j][0] = GT.Ax[j][hi][n]; av[j][1] = GT.Ay[j][hi][n]; }
    float g1r[8], g2r[8];
#pragma unroll
    for (int i = 0; i < 8; ++i) { g1r[i] = GT.g1[8 * hi + i]; g2r[i] = GT.g2[8 * hi + i]; }

    float2 st = ini[seqn];
    float y1 = st.x, y2 = st.y;

    // FIR history registers (raw x[t-2], x[t-1] preceding the next tile)
    float hx2 = (tstart >= 2) ? xcol[-2] : 0.0f;
    float hx1 = (tstart >= 1) ? xcol[-1] : 0.0f;

    // issue tile `ti` (64 steps x 16 seqs): 8 async b128 per wave,
    // lane covers its own seq `n`, half `hi` covers 32 of the 64 floats
    auto issue_tile = [&](int ti) {
        const float* gsrc = xcol + ti * TSTEP + 32 * hi;
        float* ldst = &stage[ti & 1][n][32 * hi];
#pragma unroll
        for (int j = 0; j < 8; ++j)
            ee_copy_b128(gsrc + 4 * j, ldst + 4 * j);
    };

    issue_tile(0);
    for (int ti = 0; ti < NT; ++ti) {
        const int bi = ti & 1;
        if (ti + 1 < NT) { issue_tile(ti + 1); ee_wait_async_le8(); }
        else             { ee_wait_async_0(); }
#if !EE_HAVE_ASYNC
        __syncthreads();
#endif
#pragma unroll
        for (int bq = 0; bq < 4; ++bq) {
            const int tl = bq * 16;                  // tile-local time base
            const int tb = ti * TSTEP + tl;          // chunk-local time base
            // 16 |x| values covering this lane's B rows (+FIR history)
            float X[16];
#pragma unroll
            for (int i = 0; i < 16; ++i) {
                int tt = tl + 2 * hi - 2 + i;        // index into staged tile
                float raw;
                if (tt >= 0) raw = stage[bi][n][tt];
                else         raw = (i == 0) ? hx2 : hx1;   // only bq==0, hi==0
                X[i] = fabsf(raw);
            }
            // FIR -> B operand slices (K rows 2*hi, 2*hi+1 of each K=4 group)
            v2f bv[4];
#pragma unroll
            for (int j = 0; j < 4; ++j) {
                bv[j][0] = B0f * X[4 * j + 2] + B1f * X[4 * j + 1] + B2f * X[4 * j];
                bv[j][1] = B0f * X[4 * j + 3] + B1f * X[4 * j + 2] + B2f * X[4 * j + 1];
            }
            // initial-state (homogeneous) contribution as the accumulator
            v8f acc;
#pragma unroll
            for (int i = 0; i < 8; ++i) acc[i] = g1r[i] * y1 + g2r[i] * y2;
            // y16 = Tmat(16x16) * F(16x16) + acc  — 4 chained f32 WMMAs
            acc = __builtin_amdgcn_wmma_f32_16x16x4_f32(false, av[0], false, bv[0], (short)0, acc, false, false);
            acc = __builtin_amdgcn_wmma_f32_16x16x4_f32(false, av[1], false, bv[1], (short)0, acc, false, false);
            acc = __builtin_amdgcn_wmma_f32_16x16x4_f32(false, av[2], false, bv[2], (short)0, acc, false, false);
            acc = __builtin_amdgcn_wmma_f32_16x16x4_f32(false, av[3], false, bv[3], (short)0, acc, false, false);
            // carry state = rows 14,15 (VGPR 6,7 of lanes 16..31, column n)
            float ny2 = __shfl(acc[6], 16 + n, 32);
            float ny1 = __shfl(acc[7], 16 + n, 32);
            // transpose through LDS, store two coalesced float4 per lane
#pragma unroll
            for (int i = 0; i < 8; ++i) lt[8 * hi + i][n] = acc[i];
            __syncthreads();
            float4 o0, o1;
            o0.x = lt[th + 0][sseq]; o0.y = lt[th + 1][sseq];
            o0.z = lt[th + 2][sseq]; o0.w = lt[th + 3][sseq];
            o1.x = lt[th + 4][sseq]; o1.y = lt[th + 5][sseq];
            o1.z = lt[th + 6][sseq]; o1.w = lt[th + 7][sseq];
            *reinterpret_cast<float4*>(ycol + tb + th)     = o0;
            *reinterpret_cast<float4*>(ycol + tb + th + 4) = o1;
            if (bq == 3) {
                // carry raw FIR history; the __syncthreads below (dscnt wait)
                // orders these LDS reads before the next tile's async overwrite
                hx2 = stage[bi][n][TSTEP - 2];
                hx1 = stage[bi][n][TSTEP - 1];
            }
            __syncthreads();
            y1 = ny1; y2 = ny2;
        }
    }
}

// ---------------- launcher -------------------------------------------------
extern "C" void kernel_launch(void* const* d_in, const int* in_sizes, int n_in,
                              void* d_out, int out_size, void* d_ws, size_t ws_size,
                              hipStream_t stream) {
    (void)in_sizes; (void)n_in; (void)out_size; (void)ws_size;
    const float* x = (const float*)d_in[0];
    float* out = (float*)d_out;
    float2* pst = (float2*)d_ws;          // BC float2  (384 KB)
    float2* ini = pst + BC;               // BC float2  (384 KB)

    ee_phase1<<<(BC + 127) / 128, 128, 0, stream>>>(x, pst);
    ee_phase2<<<1, 64, 0, stream>>>(pst, ini);
    ee_phase3<<<BC / 16, 32, 0, stream>>>(x, ini, out);
}